// PromptViT_43602507989355
// MI455X (gfx1250) — compile-verified
//
#include <hip/hip_runtime.h>
#include <math.h>

#define BB 64
#define CC 1000
#define NN 50000
#define DD 768
#define KSPLIT 25
#define KCHUNK (NN / KSPLIT)   // 2000, divisible by 4
#define THR_C 0.005f
#define THR_ENT 2.0f
#define ALPHA_C 0.1f

typedef __attribute__((ext_vector_type(2))) float v2f;
typedef __attribute__((ext_vector_type(8))) float v8f;

__device__ __forceinline__ v8f wmma4(v2f a, v2f b, v8f c) {
  // V_WMMA_F32_16X16X4_F32: D = A(16x4) * B(4x16) + C(16x16)
  return __builtin_amdgcn_wmma_f32_16x16x4_f32(
      /*neg_a=*/false, a, /*neg_b=*/false, b,
      /*c_mod=*/(short)0, c, /*reuse_a=*/false, /*reuse_b=*/false);
}

// ---------------------------------------------------------------------------
// K0: per-row softmax of logits -> probs, normalized probs -> pn,
//     entropy gate -> entOK.
// grid: 64 blocks (one per b), 256 threads
// ---------------------------------------------------------------------------
__global__ void __launch_bounds__(256) k_prep(const float* __restrict__ logits,
                                              float* __restrict__ probs,
                                              float* __restrict__ pn,
                                              float* __restrict__ entOK) {
  int b = blockIdx.x, t = threadIdx.x;
  __shared__ float red[256];
  const float* row = logits + (size_t)b * CC;

  float m = -1e30f;
  for (int i = t; i < CC; i += 256) m = fmaxf(m, row[i]);
  red[t] = m; __syncthreads();
  for (int s = 128; s > 0; s >>= 1) { if (t < s) red[t] = fmaxf(red[t], red[t + s]); __syncthreads(); }
  m = red[0]; __syncthreads();

  float z = 0.f, tw = 0.f, q = 0.f;
  for (int i = t; i < CC; i += 256) {
    float x = row[i] - m;
    float e = expf(x);
    z += e; tw += e * x; q += e * e;
  }
  red[t] = z; __syncthreads();
  for (int s = 128; s > 0; s >>= 1) { if (t < s) red[t] += red[t + s]; __syncthreads(); }
  z = red[0]; __syncthreads();
  red[t] = tw; __syncthreads();
  for (int s = 128; s > 0; s >>= 1) { if (t < s) red[t] += red[t + s]; __syncthreads(); }
  tw = red[0]; __syncthreads();
  red[t] = q; __syncthreads();
  for (int s = 128; s > 0; s >>= 1) { if (t < s) red[t] += red[t + s]; __syncthreads(); }
  q = red[0]; __syncthreads();

  float invZ = 1.f / z;
  float rq = rsqrtf(q);
  for (int i = t; i < CC; i += 256) {
    float e = expf(row[i] - m);
    probs[(size_t)b * CC + i] = e * invZ; // softmax
    pn[(size_t)b * CC + i]    = e * rq;   // softmax / ||softmax||
  }
  if (t == 0) {
    float ent = logf(z) - tw / z;                 // -sum p log p
    entOK[b] = (ent <= THR_ENT) ? 1.f : 0.f;
  }
}

// ---------------------------------------------------------------------------
// K1: sim = pn @ (keys/||keys||)^T via V_WMMA_F32_16X16X4_F32.
//     Writes esim[n*64+b] = (sim>THR_C) ? exp(sim) : 0.
// grid: N/16 = 3125 blocks, 128 threads (4 waves = 4 b-tiles of 16)
// ---------------------------------------------------------------------------
__global__ void __launch_bounds__(128) k_sim(const float* __restrict__ keys,
                                             const float* __restrict__ pn,
                                             float* __restrict__ esim) {
  int lane = threadIdx.x & 31;
  int wave = threadIdx.x >> 5;      // b-tile 0..3
  int ntile = blockIdx.x;
  int m16 = lane & 15;
  int hk = lane >> 4;               // half selector
  int koff = hk << 1;               // K offset 0 or 2

  int n = ntile * 16 + m16;
  const float* krow = keys + (size_t)n * CC + koff;                  // B operand rows
  const float* prow = pn + (size_t)(wave * 16 + m16) * CC + koff;    // A operand rows

  v8f c = {};
  float acc = 0.f;                   // keys row-norm^2 partial
  for (int k0 = 0; k0 < CC; k0 += 4) {
    v2f a, bm;
    a.x = prow[k0]; a.y = prow[k0 + 1];
    bm.x = krow[k0]; bm.y = krow[k0 + 1];
    acc = fmaf(bm.x, bm.x, fmaf(bm.y, bm.y, acc));
    c = wmma4(a, bm, c);
  }
  float tot = acc + __shfl_xor(acc, 16, 32);   // full ||keys[n]||^2 in every lane
  float rn = rsqrtf(tot);                       // lane's n == C-matrix column n

#pragma unroll
  for (int r = 0; r < 8; ++r) {
    int b = wave * 16 + r + (hk << 3);
    float sim = c[r] * rn;
    float e = (sim > THR_C) ? expf(sim) : 0.f;
    esim[(size_t)n * BB + b] = e;
  }
}

// ---------------------------------------------------------------------------
// K2: deterministic column reduction S[b] = sum_n esim[n,b]
// grid: 64 blocks (one per b), 256 threads; fixed order -> bitwise stable
// ---------------------------------------------------------------------------
__global__ void __launch_bounds__(256) k_ssum(const float* __restrict__ esim,
                                              float* __restrict__ S) {
  int b = blockIdx.x, t = threadIdx.x;
  __shared__ float red[256];
  float s = 0.f;
  for (int n = t; n < NN; n += 256) s += esim[(size_t)n * BB + b];
  red[t] = s; __syncthreads();
  for (int w = 128; w > 0; w >>= 1) { if (t < w) red[t] += red[t + w]; __syncthreads(); }
  if (t == 0) S[b] = red[0];
}

// ---------------------------------------------------------------------------
// K3: per-n suffix-product scan over b (B=64):
//     akT[n,b] = ALPHA*wv*Sk, apT[n,b] = wv*Sp, Pk[n]=prod fk, Pp[n]=prod fp
// grid: ceil(N/256) blocks, 256 threads
// ---------------------------------------------------------------------------
__global__ void __launch_bounds__(256) k_coef(const float* __restrict__ esim,
                                              const float* __restrict__ S,
                                              const float* __restrict__ entOK,
                                              float* __restrict__ akT,
                                              float* __restrict__ apT,
                                              float* __restrict__ Pk,
                                              float* __restrict__ Pp) {
  __shared__ float sF[BB];
  int t = threadIdx.x;
  if (t < BB) {
    float s = S[t];
    // wv[b,n] = esim[n,b] * sF[b]  (valid = matched && ent<=THR)
    sF[t] = (s > 0.f && entOK[t] > 0.5f) ? (1.f / s) : 0.f;
  }
  __syncthreads();
  int n = blockIdx.x * 256 + t;
  if (n >= NN) return;
  const float* e = esim + (size_t)n * BB;
  float* ak = akT + (size_t)n * BB;
  float* ap = apT + (size_t)n * BB;
  float sk = 1.f, sp = 1.f;
  for (int b = BB - 1; b >= 0; --b) {
    float wv = e[b] * sF[b];
    ak[b] = ALPHA_C * wv * sk;  sk *= (1.f - ALPHA_C * wv);
    ap[b] = wv * sp;            sp *= (1.f - wv);
  }
  Pk[n] = sk;
  Pp[n] = sp;
}

// ---------------------------------------------------------------------------
// K4a: partial retrieved tiles: racc[kc][b][d-tile] = esim_chunk @ pv_chunk
// grid: (D/16 = 48, KSPLIT = 25) blocks, 128 threads (4 waves = 4 b-tiles)
// ---------------------------------------------------------------------------
__global__ void __launch_bounds__(128) k_retr_part(const float* __restrict__ esim,
                                                   const float* __restrict__ pv,
                                                   float* __restrict__ racc) {
  int lane = threadIdx.x & 31;
  int wave = threadIdx.x >> 5;
  int d0 = blockIdx.x * 16;
  int kc = blockIdx.y;
  int kbase = kc * KCHUNK;
  int m16 = lane & 15;
  int hk = lane >> 4;
  int koff = hk << 1;
  int brow = wave * 16 + m16;

  v8f c = {};
  for (int k0 = kbase; k0 < kbase + KCHUNK; k0 += 4) {
    int k = k0 + koff;
    v2f a, bm;
    a.x = esim[(size_t)k * BB + brow];
    a.y = esim[(size_t)(k + 1) * BB + brow];
    bm.x = pv[(size_t)k * DD + d0 + m16];
    bm.y = pv[(size_t)(k + 1) * DD + d0 + m16];
    c = wmma4(a, bm, c);
  }
#pragma unroll
  for (int r = 0; r < 8; ++r) {
    int b = wave * 16 + r + (hk << 3);
    racc[((size_t)kc * BB + b) * DD + d0 + m16] = c[r];
  }
}

// ---------------------------------------------------------------------------
// K4b: retrieved[b,d] = matched ? (sum_kc racc)/S[b] : init_prompt[d]
// grid: (B*D)/256 = 192 blocks, 256 threads; fixed-order combine
// ---------------------------------------------------------------------------
__global__ void __launch_bounds__(256) k_retr_fin(const float* __restrict__ racc,
                                                  const float* __restrict__ S,
                                                  const float* __restrict__ ip,
                                                  float* __restrict__ retr) {
  int idx = blockIdx.x * 256 + threadIdx.x;
  if (idx >= BB * DD) return;
  int b = idx / DD;
  int d = idx - b * DD;
  float s = 0.f;
#pragma unroll
  for (int kc = 0; kc < KSPLIT; ++kc)
    s += racc[(size_t)kc * BB * DD + idx];
  float Sb = S[b];
  retr[idx] = (Sb > 0.f) ? (s / Sb) : ip[d];
}

// ---------------------------------------------------------------------------
// K5: keys_new[n,c] = Pk[n]*keys[n,c] + sum_b akT[n,b]*probs[b,c]
// grid: (N/16, 8) blocks, 256 threads (8 waves = 8 c-tiles); K = B = 64
// ---------------------------------------------------------------------------
__global__ void __launch_bounds__(256) k_keys(const float* __restrict__ keys,
                                              const float* __restrict__ probs,
                                              const float* __restrict__ akT,
                                              const float* __restrict__ Pk,
                                              float* __restrict__ kout) {
  int lane = threadIdx.x & 31;
  int wave = threadIdx.x >> 5;
  int ntile = blockIdx.x;
  int ctile = blockIdx.y * 8 + wave;
  if (ctile >= (CC + 15) / 16) return;   // wave-uniform exit
  int c0 = ctile * 16;
  int m16 = lane & 15;
  int hk = lane >> 4;
  int koff = hk << 1;
  int col = c0 + m16;
  bool cok = col < CC;
  int nrow = ntile * 16 + m16;
  const float* arow = akT + (size_t)nrow * BB;

  v8f acc = {};
#pragma unroll
  for (int kk = 0; kk < BB / 4; ++kk) {
    int k = kk * 4 + koff;
    v2f a, bm;
    a.x = arow[k]; a.y = arow[k + 1];
    bm.x = cok ? probs[(size_t)k * CC + col] : 0.f;
    bm.y = cok ? probs[(size_t)(k + 1) * CC + col] : 0.f;
    acc = wmma4(a, bm, acc);
  }
#pragma unroll
  for (int r = 0; r < 8; ++r) {
    int n = ntile * 16 + r + (hk << 3);
    if (cok)
      kout[(size_t)n * CC + col] = Pk[n] * keys[(size_t)n * CC + col] + acc[r];
  }
}

// ---------------------------------------------------------------------------
// K6: prompts_new[n,d] = Pp[n]*prompt_vals[n,d] + sum_b apT[n,b]*retrieved[b,d]
// grid: (N/16, 6) blocks, 256 threads (8 waves = 8 d-tiles, 48 total); K = 64
// ---------------------------------------------------------------------------
__global__ void __launch_bounds__(256) k_prompts(const float* __restrict__ pv,
                                                 const float* __restrict__ retr,
                                                 const float* __restrict__ apT,
                                                 const float* __restrict__ Pp,
                                                 float* __restrict__ pout) {
  int lane = threadIdx.x & 31;
  int wave = threadIdx.x >> 5;
  int ntile = blockIdx.x;
  int dtile = blockIdx.y * 8 + wave;     // 0..47, always valid
  int d0 = dtile * 16;
  int m16 = lane & 15;
  int hk = lane >> 4;
  int koff = hk << 1;
  int col = d0 + m16;
  int nrow = ntile * 16 + m16;
  const float* arow = apT + (size_t)nrow * BB;

  v8f acc = {};
#pragma unroll
  for (int kk = 0; kk < BB / 4; ++kk) {
    int k = kk * 4 + koff;
    v2f a, bm;
    a.x = arow[k]; a.y = arow[k + 1];
    bm.x = retr[(size_t)k * DD + col];
    bm.y = retr[(size_t)(k + 1) * DD + col];
    acc = wmma4(a, bm, acc);
  }
#pragma unroll
  for (int r = 0; r < 8; ++r) {
    int n = ntile * 16 + r + (hk << 3);
    pout[(size_t)n * DD + col] = Pp[n] * pv[(size_t)n * DD + col] + acc[r];
  }
}

// ---------------------------------------------------------------------------
extern "C" void kernel_launch(void* const* d_in, const int* in_sizes, int n_in,
                              void* d_out, int out_size, void* d_ws, size_t ws_size,
                              hipStream_t stream) {
  (void)in_sizes; (void)n_in; (void)out_size; (void)ws_size;
  const float* logits = (const float*)d_in[0];   // (64,1000)
  const float* keys   = (const float*)d_in[1];   // (50000,1000)
  const float* pv     = (const float*)d_in[2];   // (50000,1,768)
  const float* ip     = (const float*)d_in[3];   // (1,1,768)

  float* out  = (float*)d_out;
  float* retr = out;                                   // 64*768
  float* kout = out + (size_t)BB * DD;                 // 50000*1000
  float* pout = kout + (size_t)NN * CC;                // 50000*768

  float* ws    = (float*)d_ws;
  float* esim  = ws;                                   // N*B
  float* probs = esim + (size_t)NN * BB;               // B*C
  float* pn    = probs + (size_t)BB * CC;              // B*C
  float* S     = pn + (size_t)BB * CC;                 // B
  float* entOK = S + BB;                               // B
  float* akT   = entOK + BB;                           // N*B
  float* apT   = akT + (size_t)NN * BB;                // N*B
  float* Pk    = apT + (size_t)NN * BB;                // N
  float* Pp    = Pk + NN;                              // N
  float* racc  = Pp + NN;                              // KSPLIT*B*D (~4.9MB)

  k_prep<<<BB, 256, 0, stream>>>(logits, probs, pn, entOK);
  k_sim<<<NN / 16, 128, 0, stream>>>(keys, pn, esim);
  k_ssum<<<BB, 256, 0, stream>>>(esim, S);
  k_coef<<<(NN + 255) / 256, 256, 0, stream>>>(esim, S, entOK, akT, apT, Pk, Pp);
  k_retr_part<<<dim3(DD / 16, KSPLIT), 128, 0, stream>>>(esim, pv, racc);
  k_retr_fin<<<(BB * DD + 255) / 256, 256, 0, stream>>>(racc, S, ip, retr);
  k_keys<<<dim3(NN / 16, 8), 256, 0, stream>>>(keys, probs, akT, Pk, kout);
  k_prompts<<<dim3(NN / 16, 6), 256, 0, stream>>>(pv, retr, apT, Pp, pout);
}